// EigLayer_36996848287831
// MI455X (gfx1250) — compile-verified
//
#include <hip/hip_runtime.h>
#include <hip/hip_bf16.h>

// Batched symmetric eigendecomposition (jnp.linalg.eigh equivalent) for
// N=8192 matrices of size 64x64, fp32.
//
// Strategy (MI455X / gfx1250):
//  - One workgroup (128 threads = 4 wave32) per matrix; A, Q, U live in LDS
//    (~50KB, well under the 320KB/WGP budget).
//  - Cyclic two-sided Jacobi: 63 round-robin rounds per sweep, 32 disjoint
//    pivot pairs per round. Rotation application to A and accumulation into
//    the per-sweep rotation product Q are scalar VALU ops on LDS
//    (bank-conflict-free via row stride 65).
//  - Eigenvector accumulation U <- U*Q once per sweep as a dense 64x64x64
//    fp32 GEMM on the matrix cores via V_WMMA_F32_16X16X4_F32 (exact fp32,
//    1024 MACs/instruction/wave vs 32 for v_fma_f32).
//  - Final rank-sort reproduces eigh's ascending eigenvalue order; output is
//    S = diag(w) ([N,64,64]) followed by U ([N,64,64]).

#define NMAT    8192
#define D       64
#define PAD     65          // LDS row stride in floats: conflict-free rows AND columns
#define NSWEEPS 8
#define THREADS 128

typedef __attribute__((ext_vector_type(2))) float v2f;
typedef __attribute__((ext_vector_type(8))) float v8f;

__global__ __launch_bounds__(THREADS)
void eigh_jacobi_wmma_kernel(const float* __restrict__ x,
                             float* __restrict__ outS,
                             float* __restrict__ outU)
{
    __shared__ float A[D][PAD];
    __shared__ float Q[D][PAD];
    __shared__ float U[D][PAD];
    __shared__ float cs_c[32];
    __shared__ float cs_s[32];
    __shared__ int   pArr[32];
    __shared__ int   qArr[32];
    __shared__ float wvals[D];
    __shared__ float wsorted[D];
    __shared__ int   colmap[D];

    const int tid  = threadIdx.x;
    const int lane = tid & 31;
    const int wv   = tid >> 5;          // wave id 0..3
    const int half = lane >> 4;         // 0: lanes 0-15, 1: lanes 16-31
    const int ml   = lane & 15;

    const int mat = blockIdx.x;
    const float* xin = x + (size_t)mat * (D * D);

    // ---- load A, init U = I ----
    for (int idx = tid; idx < D * D; idx += THREADS) {
        int r = idx >> 6, c = idx & 63;
        A[r][c] = xin[idx];
        U[r][c] = (r == c) ? 1.0f : 0.0f;
    }
    __syncthreads();

    for (int sweep = 0; sweep < NSWEEPS; ++sweep) {
        // ---- Q = I ----
        for (int idx = tid; idx < D * D; idx += THREADS) {
            int r = idx >> 6, c = idx & 63;
            Q[r][c] = (r == c) ? 1.0f : 0.0f;
        }
        __syncthreads();

        // ---- 63 round-robin rounds of disjoint Jacobi rotations ----
        for (int rnd = 0; rnd < 63; ++rnd) {
            // one thread per pivot pair: compute rotation (c,s)
            if (tid < 32) {
                int p, q;
                if (tid == 0) { p = 63; q = rnd; }
                else {
                    p = (rnd + tid) % 63;
                    q = (rnd + 63 - tid) % 63;
                }
                float app = A[p][p];
                float aqq = A[q][q];
                float apq = A[p][q];
                float c_, s_;
                if (fabsf(apq) < 1e-20f) {
                    c_ = 1.0f; s_ = 0.0f;
                } else {
                    float tau = (aqq - app) / (2.0f * apq);
                    float t = ((tau >= 0.0f) ? 1.0f : -1.0f) /
                              (fabsf(tau) + sqrtf(1.0f + tau * tau));
                    c_ = 1.0f / sqrtf(1.0f + t * t);
                    s_ = t * c_;
                }
                cs_c[tid] = c_;
                cs_s[tid] = s_;
                pArr[tid] = p;
                qArr[tid] = q;
            }
            __syncthreads();

            // row updates: A' = J^T * A (32 pairs x 64 cols = 2048 items)
            for (int it = 0; it < 16; ++it) {
                int idx = it * THREADS + tid;     // lanes walk k contiguously
                int j = idx >> 6;
                int k = idx & 63;
                int p = pArr[j], q = qArr[j];
                float c_ = cs_c[j], s_ = cs_s[j];
                float ap = A[p][k], aq = A[q][k];
                A[p][k] =  c_ * ap + s_ * aq;
                A[q][k] = -s_ * ap + c_ * aq;
            }
            __syncthreads();

            // column updates: A'' = A' * J, and accumulate Q = Q * J
            for (int it = 0; it < 16; ++it) {
                int idx = it * THREADS + tid;
                int j = idx >> 6;
                int k = idx & 63;
                int p = pArr[j], q = qArr[j];
                float c_ = cs_c[j], s_ = cs_s[j];

                float ap = A[k][p], aq = A[k][q];
                A[k][p] =  c_ * ap + s_ * aq;
                A[k][q] = -s_ * ap + c_ * aq;

                float up = Q[k][p], uq = Q[k][q];
                Q[k][p] =  c_ * up + s_ * uq;
                Q[k][q] = -s_ * up + c_ * uq;
            }
            __syncthreads();
        }

        // ---- U <- U * Q : dense 64x64x64 fp32 GEMM on matrix cores ----
        // wave wv computes output rows [wv*16, wv*16+16), all 4 column tiles.
        v8f acc[4];
        const int ti = wv;
        for (int tj = 0; tj < 4; ++tj) {
            v8f c = {0.f, 0.f, 0.f, 0.f, 0.f, 0.f, 0.f, 0.f};
            for (int k0 = 0; k0 < D; k0 += 4) {
                // A-frag (16x4 fp32): lanes 0-15 hold K={k0,k0+1},
                //                     lanes 16-31 hold K={k0+2,k0+3}
                v2f a, b;
                int m = ti * 16 + ml;
                a[0] = U[m][k0 + 2 * half + 0];
                a[1] = U[m][k0 + 2 * half + 1];
                // B-frag (4x16 fp32), mirrored layout
                int n = tj * 16 + ml;
                b[0] = Q[k0 + 2 * half + 0][n];
                b[1] = Q[k0 + 2 * half + 1][n];
                c = __builtin_amdgcn_wmma_f32_16x16x4_f32(
                        /*neg_a=*/false, a, /*neg_b=*/false, b,
                        /*c_mod=*/(short)0, c,
                        /*reuse_a=*/false, /*reuse_b=*/false);
            }
            acc[tj] = c;
        }
        __syncthreads();   // all reads of U done before overwriting
        for (int tj = 0; tj < 4; ++tj) {
            for (int g = 0; g < 8; ++g) {
                int m = ti * 16 + g + 8 * half;   // C/D layout: VGPR g -> M=g / g+8
                int n = tj * 16 + ml;
                U[m][n] = acc[tj][g];
            }
        }
        __syncthreads();
    }

    // ---- eigenvalues = diag(A); rank-sort ascending (eigh convention) ----
    if (tid < D) wvals[tid] = A[tid][tid];
    __syncthreads();
    if (tid < D) {
        float wi = wvals[tid];
        int rank = 0;
        for (int j = 0; j < D; ++j) {
            float wj = wvals[j];
            rank += (wj < wi) || (wj == wi && j < tid);
        }
        colmap[tid] = rank;
    }
    __syncthreads();
    if (tid < D) wsorted[colmap[tid]] = wvals[tid];
    __syncthreads();

    // ---- write outputs ----
    float* So = outS + (size_t)mat * (D * D);
    float* Uo = outU + (size_t)mat * (D * D);
    for (int idx = tid; idx < D * D; idx += THREADS) {
        int r = idx >> 6, c = idx & 63;
        So[idx] = (r == c) ? wsorted[r] : 0.0f;
    }
    for (int idx = tid; idx < D * D; idx += THREADS) {
        int r = idx >> 6, i = idx & 63;     // i = original column of U
        Uo[r * D + colmap[i]] = U[r][i];
    }
}

extern "C" void kernel_launch(void* const* d_in, const int* in_sizes, int n_in,
                              void* d_out, int out_size, void* d_ws, size_t ws_size,
                              hipStream_t stream) {
    (void)in_sizes; (void)n_in; (void)d_ws; (void)ws_size; (void)out_size;

    const float* x = (const float*)d_in[0];
    float* out  = (float*)d_out;
    float* outS = out;                                  // [N,64,64]
    float* outU = out + (size_t)NMAT * D * D;           // [N,64,64]

    eigh_jacobi_wmma_kernel<<<dim3(NMAT), dim3(THREADS), 0, stream>>>(x, outS, outU);
}